// SemiMarkovCRFHead_18562848653916
// MI455X (gfx1250) — compile-verified
//
#include <hip/hip_runtime.h>
#include <hip/hip_bf16.h>

#define NEGF (-1e30f)

#define Bc 8
#define Tc 16384
#define Hc 512
#define Cc 24
#define Kc 100

typedef __attribute__((ext_vector_type(2))) float v2f;
typedef __attribute__((ext_vector_type(8))) float v8f;

// ---------------------------------------------------------------------------
// Kernel 1: scores[b*T+t, c] = hs[b*T+t, :] @ W[:, c] + bias[c]
// f32 WMMA 16x16x4, one wave computes a 16-row x 32-col (2 N-tiles) strip,
// accumulating over K=512 in steps of 4. Pure HBM-bound (~12us roofline on
// 256 MB of hs at 23.3 TB/s), so exact-precision f32 WMMA is the right pick.
// ---------------------------------------------------------------------------
__global__ __launch_bounds__(128)
void crf_scores_wmma(const float* __restrict__ hs, const float* __restrict__ W,
                     const float* __restrict__ bias, float* __restrict__ scores) {
  const int lane = threadIdx.x & 31;
  const int wave = threadIdx.x >> 5;
  const int half = lane >> 4;   // 0: lanes 0-15, 1: lanes 16-31
  const int l15  = lane & 15;
  const long row0 = ((long)blockIdx.x * 4 + wave) * 16;

  // B-fragment columns: tile0 = 0..15 (always valid), tile1 = 16..31 (valid < 24)
  const int   n0    = l15;
  const int   n1raw = 16 + l15;
  const float m1    = (n1raw < Cc) ? 1.0f : 0.0f;   // arithmetic mask, no branch
  const int   n1    = (n1raw < Cc) ? n1raw : 0;

  // A layout (ISA 7.12.2, 32-bit A 16x4): lanes 0-15 hold K=kk,kk+1 for row l15,
  // lanes 16-31 hold K=kk+2,kk+3 for row l15.
  const float* arow = hs + (row0 + l15) * Hc;

  v8f acc0 = {};
  v8f acc1 = {};
  for (int kk = 0; kk < Hc; kk += 4) {
    // hide HBM latency on the streaming A read; locality=3 -> near (WGP) scope,
    // pulls into all cache levels (bounds-guarded: WGP prefetch is non-speculative)
    if (kk + 64 < Hc) __builtin_prefetch(arow + kk + 64, 0, 3);
    const int ka = kk + 2 * half;
    v2f a;
    a.x = arow[ka + 0];
    a.y = arow[ka + 1];
    v2f b0, b1;
    b0.x = W[(ka + 0) * Cc + n0];
    b0.y = W[(ka + 1) * Cc + n0];
    b1.x = W[(ka + 0) * Cc + n1] * m1;
    b1.y = W[(ka + 1) * Cc + n1] * m1;
    // 8 args: (neg_a, A, neg_b, B, c_mod, C, reuse_a, reuse_b)
    acc0 = __builtin_amdgcn_wmma_f32_16x16x4_f32(false, a, false, b0, (short)0,
                                                 acc0, false, false);
    acc1 = __builtin_amdgcn_wmma_f32_16x16x4_f32(false, a, false, b1, (short)0,
                                                 acc1, false, false);
  }

  // D layout: VGPR v -> M = v (lanes 0-15) or 8+v (lanes 16-31); N = l15 (+16)
  const float bias0 = bias[n0];
  const float bias1 = bias[n1];   // n1 clamped; store guarded below
#pragma unroll
  for (int v = 0; v < 8; ++v) {
    const long r = row0 + (half ? (8 + v) : v);
    scores[r * Cc + n0] = acc0[v] + bias0;
    if (n1raw < Cc) scores[r * Cc + n1raw] = acc1[v] + bias1;
  }
}

// ---------------------------------------------------------------------------
// Kernel 2: streaming semi-Markov CRF forward. One workgroup per batch,
// sequential over t with all O(K*C) ring state LDS-resident.
//   msgRing[slot(s)][c]  = msg at time s,   slot(s) = (s+K) mod K = (t+k) mod K
//   cumRing[slot(s)][c]  = cum[s][c]
// init: msg[0]=0, msg[s<0]=NEG (slots not yet overwritten hold init values).
// 3 barriers per step: the cum update for step t+1 is folded into the tail of
// step t's channel-thread phase (same owner thread; end-of-loop barrier orders
// it against next step's window readers).
// ---------------------------------------------------------------------------
__global__ __launch_bounds__(256)
void crf_forward(const float* __restrict__ scores, const float* __restrict__ transition,
                 const float* __restrict__ duration_bias, const int* __restrict__ lengths,
                 float* __restrict__ out) {
  __shared__ float msgRing[Kc * Cc];
  __shared__ float cumRing[Kc * Cc];
  __shared__ float durS[Kc * Cc];
  __shared__ float transS[Cc * Cc];
  __shared__ float curCum[Cc];
  __shared__ float alphaS[Cc];
  __shared__ float pm[10 * Cc];
  __shared__ float ps[10 * Cc];

  const int tid = threadIdx.x;
  const int b = blockIdx.x;
  int Lb = lengths[b];
  if (Lb < 1) Lb = 1;
  if (Lb > Tc) Lb = Tc;

  const float* sb = scores + (size_t)b * Tc * Cc;

  for (int i = tid; i < Kc * Cc; i += 256) {
    msgRing[i] = (i < Cc) ? 0.0f : NEGF;  // slot 0 = msg[s=0] = 0
    cumRing[i] = 0.0f;                    // cum[s<=0] = 0
    durS[i] = duration_bias[i];
  }
  for (int i = tid; i < Cc * Cc; i += 256) transS[i] = transition[i];
  if (tid < Cc) curCum[tid] = sb[tid];    // cum[1] = scores[b,0,:]
  __syncthreads();

  const int c  = tid % Cc;   // channel
  const int sl = tid / Cc;   // k-slice 0..9 (tid < 240 active in phase 2)

  for (int t = 1; t <= Tc; ++t) {
    // ---- P2: partial streaming LSE over window slices (10 k-values/thread)
    // curCum currently holds cum[t].
    if (tid < 240) {
      int slot = (t + sl * 10) % Kc;      // slot for s = t-K + sl*10
      int drow = Kc - 1 - sl * 10;        // duration index k-1 = K-1-k_idx
      const float ct = curCum[c];
      float m = NEGF, ssum = 0.0f;
#pragma unroll
      for (int j = 0; j < 10; ++j) {
        const float val = msgRing[slot * Cc + c] + (ct - cumRing[slot * Cc + c])
                        + durS[drow * Cc + c];
        if (val > m) { ssum = ssum * __expf(m - val) + 1.0f; m = val; }
        else         { ssum += __expf(val - m); }
        ++slot; if (slot == Kc) slot = 0;
        --drow;
      }
      pm[sl * Cc + c] = m;
      ps[sl * Cc + c] = ssum;
    } else if (tid == 255) {
      // prefetch scores rows ~16 steps ahead into near caches (L2-resident data;
      // near-scope prefetch hides L2->WGP latency; bounds-guarded)
      if (t + 16 <= Tc) __builtin_prefetch(&sb[(size_t)(t + 15) * Cc], 0, 3);
    }
    __syncthreads();

    // ---- P3: combine slice partials -> alpha_t[c]
    if (tid < Cc) {
      float m = NEGF, ssum = 0.0f;
#pragma unroll
      for (int s2 = 0; s2 < 10; ++s2) {
        const float mm = pm[s2 * Cc + tid];
        const float ss = ps[s2 * Cc + tid];
        if (mm > m) { ssum = ssum * __expf(m - mm) + ss; m = mm; }
        else        { ssum += ss * __expf(mm - m); }
      }
      alphaS[tid] = m + __logf(ssum);
    }
    __syncthreads();

    // ---- P4: msg_t[c] = LSE_{c'} alpha[c'] + trans[c'][c]; ring update;
    //          then fold in the cum update for step t+1.
    if (tid < Cc) {
      float m = NEGF, ssum = 0.0f;
#pragma unroll
      for (int cp = 0; cp < Cc; ++cp) {
        const float val = alphaS[cp] + transS[cp * Cc + tid];
        if (val > m) { ssum = ssum * __expf(m - val) + 1.0f; m = val; }
        else         { ssum += __expf(val - m); }
      }
      const int wslot = t % Kc;
      msgRing[wslot * Cc + tid] = m + __logf(ssum);
      cumRing[wslot * Cc + tid] = curCum[tid];      // cum[t] into the ring
      if (t < Tc) curCum[tid] += sb[(size_t)t * Cc + tid];  // -> cum[t+1]
    }
    // partition at t == clamp(length): alphas[L-1] == alpha at step t=L
    if (t == Lb && tid == 0) {
      float m = NEGF, ssum = 0.0f;
      for (int cp = 0; cp < Cc; ++cp) {
        const float val = alphaS[cp];
        if (val > m) { ssum = ssum * __expf(m - val) + 1.0f; m = val; }
        else         { ssum += __expf(val - m); }
      }
      out[b] = m + __logf(ssum);
    }
    __syncthreads();
    if (t == Lb) break;  // uniform across the block
  }
}

// ---------------------------------------------------------------------------
extern "C" void kernel_launch(void* const* d_in, const int* in_sizes, int n_in,
                              void* d_out, int out_size, void* d_ws, size_t ws_size,
                              hipStream_t stream) {
  const float* hs      = (const float*)d_in[0];  // (B,T,H)
  const float* W       = (const float*)d_in[1];  // (H,C)
  const float* bias    = (const float*)d_in[2];  // (C,)
  const float* trans   = (const float*)d_in[3];  // (C,C)
  const float* dur     = (const float*)d_in[4];  // (K,C)
  const int*   lengths = (const int*)d_in[5];    // (B,)
  float* out    = (float*)d_out;                 // (B,)
  float* scores = (float*)d_ws;                  // B*T*C floats = 12.6 MB scratch

  const int rowTiles = (Bc * Tc) / 16;           // 8192 16-row tiles
  crf_scores_wmma<<<rowTiles / 4, 128, 0, stream>>>(hs, W, bias, scores);
  crf_forward<<<Bc, 256, 0, stream>>>(scores, trans, dur, lengths, out);

  (void)in_sizes; (void)n_in; (void)out_size; (void)ws_size;
}